// GraphConvModel_82875688944202
// MI455X (gfx1250) — compile-verified
//
#include <hip/hip_runtime.h>
#include <math.h>

#define N_NODES 100000
#define N_EDGES 1600000
#define N_GRAPHS 2048
#define F_IN 33
#define D 256
#define LDSTRIDE 260   // 256 + 4 pad: bank step 4 per row -> conflict-free column reads

typedef float v2f __attribute__((ext_vector_type(2)));
typedef float v8f __attribute__((ext_vector_type(8)));

__device__ __forceinline__ float gelu_tanh(float x) {
    const float c = 0.7978845608028654f;   // sqrt(2/pi)
    float x3 = x * x * x;
    return 0.5f * x * (1.0f + tanhf(c * (x + 0.044715f * x3)));
}

// ---------------- zero fill ----------------
__global__ void zero_f32(float* __restrict__ p, int n) {
    int i = blockIdx.x * blockDim.x + threadIdx.x;
    if (i < n) p[i] = 0.0f;
}

// ---------------- 256x256 transpose (weights -> WT[n][k]) ----------------
__global__ __launch_bounds__(256) void transpose_256(
    const float* __restrict__ w, float* __restrict__ wt) {
    __shared__ float tile[32][33];
    int bx = blockIdx.x;            // source col tile
    int by = blockIdx.y;            // source row tile
    int tx = threadIdx.x & 31;
    int ty = threadIdx.x >> 5;      // 0..7
#pragma unroll
    for (int i = 0; i < 32; i += 8)
        tile[ty + i][tx] = w[(by * 32 + ty + i) * D + bx * 32 + tx];
    __syncthreads();
#pragma unroll
    for (int i = 0; i < 32; i += 8)
        wt[(bx * 32 + ty + i) * D + by * 32 + tx] = tile[tx][ty + i];
}

// ---------------- edge scatter-add, F = 33 (layer 0) ----------------
__global__ __launch_bounds__(64) void scatter_add_33(
    const float* __restrict__ x, const int* __restrict__ src,
    const int* __restrict__ dst, float* __restrict__ agg) {
    int e = blockIdx.x;
    int f = threadIdx.x;
    if (f >= F_IN) return;
    int s = src[e];
    int d = dst[e];
    atomicAdd(&agg[d * F_IN + f], x[s * F_IN + f]);
}

// ---------------- edge scatter-add, F = 256 ----------------
// one block (256 threads) per edge; gather of h[src] row fully coalesced (1KB)
__global__ __launch_bounds__(256) void scatter_add_256(
    const float* __restrict__ h, const int* __restrict__ src,
    const int* __restrict__ dst, float* __restrict__ agg) {
    int e = blockIdx.x;
    int f = threadIdx.x;
    int s = src[e];
    int d = dst[e];
    atomicAdd(&agg[d * D + f], h[s * D + f]);
}

// ---------------- layer 0 GEMM: K=33, VALU (negligible FLOPs) ----------------
__global__ __launch_bounds__(256) void gemm33_gelu(
    const float* __restrict__ agg, const float* __restrict__ x,
    const float* __restrict__ w_rel, const float* __restrict__ b_rel,
    const float* __restrict__ w_root, float* __restrict__ out) {
    int n = blockIdx.x;
    int d = threadIdx.x;
    const float* ar = agg + (size_t)n * F_IN;
    const float* xr = x + (size_t)n * F_IN;
    float s = b_rel[d];
#pragma unroll
    for (int k = 0; k < F_IN; ++k) {
        s += ar[k] * w_rel[k * D + d];      // ar[k]: broadcast across lanes
        s += xr[k] * w_root[k * D + d];     // weight loads coalesced across d
    }
    out[(size_t)n * D + d] = gelu_tanh(s);
}

// ---- fused dual-GEMM + bias + GELU via V_WMMA_F32_16X16X4_F32 ----
// Block owns 32 nodes (two 16-row M-tiles); wave w owns columns [w*64, w*64+64).
// Weights are pre-transposed (WT[n][k]) so a B-fragment is one b64 load,
// and every B-fragment register is reused across both M-tiles (16 WMMA : 8 weight loads).
__global__ __launch_bounds__(128) void gemm_wmma_gelu(
    const float* __restrict__ agg, const float* __restrict__ h,
    const float* __restrict__ w_relT, const float* __restrict__ b_rel,
    const float* __restrict__ w_rootT, float* __restrict__ out, int n_nodes) {
    __shared__ float sAgg[32 * LDSTRIDE];
    __shared__ float sH[32 * LDSTRIDE];

    int block_row = blockIdx.x * 32;
    int tid = threadIdx.x;

    // stage 32x256 tiles of agg and h (float4, coalesced); pad OOB rows with 0
    for (int i = tid; i < 32 * D / 4; i += 128) {
        int c = (i * 4) & (D - 1);
        int r = (i * 4) >> 8;
        int row = block_row + r;
        float4 va = make_float4(0.f, 0.f, 0.f, 0.f);
        float4 vh = va;
        if (row < n_nodes) {
            va = *(const float4*)(agg + (size_t)row * D + c);
            vh = *(const float4*)(h + (size_t)row * D + c);
        }
        *(float4*)(sAgg + r * LDSTRIDE + c) = va;
        *(float4*)(sH + r * LDSTRIDE + c) = vh;
    }
    __syncthreads();

    int wave = tid >> 5;
    int lane = tid & 31;
    int m = lane & 15;                 // A: row; B: column (within tile)
    int kb = (lane >> 4) << 1;         // K sub-offset: 0 for lanes 0-15, 2 for 16-31
    int col_base = wave * 64;

    v8f accA[4] = {};                  // rows block_row + 0..15
    v8f accB[4] = {};                  // rows block_row + 16..31

    for (int k = 0; k < D; k += 4) {
        // A fragments for both M-tiles (reused across the wave's 4 N-tiles)
        v2f aAgg0, aAgg1, aH0, aH1;
        aAgg0.x = sAgg[m * LDSTRIDE + k + kb];
        aAgg0.y = sAgg[m * LDSTRIDE + k + kb + 1];
        aAgg1.x = sAgg[(m + 16) * LDSTRIDE + k + kb];
        aAgg1.y = sAgg[(m + 16) * LDSTRIDE + k + kb + 1];
        aH0.x = sH[m * LDSTRIDE + k + kb];
        aH0.y = sH[m * LDSTRIDE + k + kb + 1];
        aH1.x = sH[(m + 16) * LDSTRIDE + k + kb];
        aH1.y = sH[(m + 16) * LDSTRIDE + k + kb + 1];

#pragma unroll
        for (int t = 0; t < 4; ++t) {
            int col = col_base + t * 16 + m;
            // contiguous b64 loads from transposed weights
            v2f bRel  = *(const v2f*)(w_relT  + (size_t)col * D + k + kb);
            v2f bRoot = *(const v2f*)(w_rootT + (size_t)col * D + k + kb);
            accA[t] = __builtin_amdgcn_wmma_f32_16x16x4_f32(
                false, aAgg0, false, bRel, (short)0, accA[t], false, false);
            accA[t] = __builtin_amdgcn_wmma_f32_16x16x4_f32(
                false, aH0, false, bRoot, (short)0, accA[t], false, false);
            accB[t] = __builtin_amdgcn_wmma_f32_16x16x4_f32(
                false, aAgg1, false, bRel, (short)0, accB[t], false, false);
            accB[t] = __builtin_amdgcn_wmma_f32_16x16x4_f32(
                false, aH1, false, bRoot, (short)0, accB[t], false, false);
        }
    }

    // epilogue: bias (per-lane scalar: N = lane&15) + GELU + store
    int row_off = (lane >> 4) * 8;
#pragma unroll
    for (int t = 0; t < 4; ++t) {
        int col = col_base + t * 16 + m;
        float bias = b_rel[col];
#pragma unroll
        for (int i = 0; i < 8; ++i) {
            int row0 = block_row + row_off + i;
            int row1 = row0 + 16;
            if (row0 < n_nodes)
                out[(size_t)row0 * D + col] = gelu_tanh(accA[t][i] + bias);
            if (row1 < n_nodes)
                out[(size_t)row1 * D + col] = gelu_tanh(accB[t][i] + bias);
        }
    }
}

// ---------------- global mean pool: scatter ----------------
__global__ __launch_bounds__(256) void pool_scatter(
    const float* __restrict__ h, const int* __restrict__ batch,
    float* __restrict__ pool, float* __restrict__ counts) {
    int n = blockIdx.x;
    int d = threadIdx.x;
    int b = batch[n];
    atomicAdd(&pool[(size_t)b * D + d], h[(size_t)n * D + d]);
    if (d == 0) atomicAdd(&counts[b], 1.0f);
}

// ---------------- readout: (pool/count) @ w_out + b_out ----------------
__global__ __launch_bounds__(256) void readout(
    const float* __restrict__ pool, const float* __restrict__ counts,
    const float* __restrict__ w_out, const float* __restrict__ b_out,
    float* __restrict__ out) {
    __shared__ float red[256];
    int g = blockIdx.x;
    int t = threadIdx.x;
    red[t] = pool[(size_t)g * D + t] * w_out[t];
    __syncthreads();
    for (int s = 128; s > 0; s >>= 1) {
        if (t < s) red[t] += red[t + s];
        __syncthreads();
    }
    if (t == 0) out[g] = red[0] / fmaxf(counts[g], 1.0f) + b_out[0];
}

extern "C" void kernel_launch(void* const* d_in, const int* in_sizes, int n_in,
                              void* d_out, int out_size, void* d_ws, size_t ws_size,
                              hipStream_t stream) {
    const float* x = (const float*)d_in[0];
    const int* edge_index = (const int*)d_in[1];
    const int* batch_index = (const int*)d_in[2];
    const float* w_rel0 = (const float*)d_in[3];
    const float* b_rel0 = (const float*)d_in[4];
    const float* w_root0 = (const float*)d_in[5];
    const float* w_rel[4] = {w_rel0, (const float*)d_in[6], (const float*)d_in[9], (const float*)d_in[12]};
    const float* b_rel[4] = {b_rel0, (const float*)d_in[7], (const float*)d_in[10], (const float*)d_in[13]};
    const float* w_root[4] = {w_root0, (const float*)d_in[8], (const float*)d_in[11], (const float*)d_in[14]};
    const float* w_out = (const float*)d_in[15];
    const float* b_out = (const float*)d_in[16];
    float* out = (float*)d_out;

    const int* src = edge_index;             // edge_index[0]
    const int* dst = edge_index + N_EDGES;   // edge_index[1]

    // workspace layout
    char* ws = (char*)d_ws;
    const size_t nd_bytes = (size_t)N_NODES * D * sizeof(float);
    const size_t w_bytes = (size_t)D * D * sizeof(float);
    float* agg = (float*)ws;                                  ws += nd_bytes;
    float* hA  = (float*)ws;                                  ws += nd_bytes;
    float* hB  = (float*)ws;                                  ws += nd_bytes;
    float* pool = (float*)ws;                                 ws += (size_t)N_GRAPHS * D * sizeof(float);
    float* counts = (float*)ws;                               ws += 4096;  // 4096 B, less than N_GRAPHS floats (8192 B)
    float* w_relT[4] = {nullptr, nullptr, nullptr, nullptr};
    float* w_rootT[4] = {nullptr, nullptr, nullptr, nullptr};
    for (int l = 1; l <= 3; ++l) {
        w_relT[l]  = (float*)ws;  ws += w_bytes;
        w_rootT[l] = (float*)ws;  ws += w_bytes;
    }

    // ---- pre-transpose layer 1..3 weights (tiny; enables b64 B-fragment loads) ----
    {
        dim3 tg(8, 8);
        for (int l = 1; l <= 3; ++l) {
            transpose_256<<<tg, 256, 0, stream>>>(w_rel[l], w_relT[l]);
            transpose_256<<<tg, 256, 0, stream>>>(w_root[l], w_rootT[l]);
        }
    }

    const int gemm_blocks = (N_NODES + 31) / 32;

    // ---- layer 0: GraphConv(F_IN -> D) ----
    {
        int nz = N_NODES * F_IN;
        zero_f32<<<(nz + 255) / 256, 256, 0, stream>>>(agg, nz);
        scatter_add_33<<<N_EDGES, 64, 0, stream>>>(x, src, dst, agg);
        gemm33_gelu<<<N_NODES, 256, 0, stream>>>(agg, x, w_rel[0], b_rel[0], w_root[0], hA);
    }

    // ---- layers 1..3: GraphConv(D -> D), WMMA ----
    float* hcur = hA;
    float* hnxt = hB;
    for (int l = 1; l <= 3; ++l) {
        int nz = N_NODES * D;
        zero_f32<<<(nz + 255) / 256, 256, 0, stream>>>(agg, nz);
        scatter_add_256<<<N_EDGES, 256, 0, stream>>>(hcur, src, dst, agg);
        gemm_wmma_gelu<<<gemm_blocks, 128, 0, stream>>>(
            agg, hcur, w_relT[l], b_rel[l], w_rootT[l], hnxt, N_NODES);
        float* t = hcur; hcur = hnxt; hnxt = t;
    }

    // ---- global mean pool + output head ----
    {
        int nz = N_GRAPHS * D;
        zero_f32<<<(nz + 255) / 256, 256, 0, stream>>>(pool, nz);
        zero_f32<<<(N_GRAPHS + 255) / 256, 256, 0, stream>>>(counts, N_GRAPHS);
        pool_scatter<<<N_NODES, 256, 0, stream>>>(hcur, batch_index, pool, counts);
        readout<<<N_GRAPHS, 256, 0, stream>>>(pool, counts, w_out, b_out, out);
    }
}